// HybridModel_38448547233945
// MI455X (gfx1250) — compile-verified
//
#include <hip/hip_runtime.h>
#include <hip/hip_bf16.h>
#include <math.h>

// ---------------- types for WMMA ----------------
typedef __attribute__((ext_vector_type(2))) float v2f;
typedef __attribute__((ext_vector_type(8))) float v8f;

#define IMG 100
#define NPS 50            // patches per side
#define NPATCH (NPS*NPS)  // 2500
#define BATCH 64
#define FEAT 10000        // NPATCH*4
#define KSPLIT 25         // K-dimension split for the WMMA GEMM
#define KCHUNK (FEAT / KSPLIT)  // 400, divisible by 8

// ---------------- 4-qubit gate helpers (compile-time strides keep the
// 16-amplitude state fully in VGPRs) ----------------
template<int S>
__device__ __forceinline__ void apply_rx(float* re, float* im, float c, float s) {
#pragma unroll
  for (int idx = 0; idx < 16; ++idx) {
    if ((idx & S) == 0) {
      const int j = idx | S;
      float r0 = re[idx], i0 = im[idx], r1 = re[j], i1 = im[j];
      re[idx] =  c * r0 + s * i1;
      im[idx] =  c * i0 - s * r1;
      re[j]   =  s * i0 + c * r1;
      im[j]   = -s * r0 + c * i1;
    }
  }
}

template<int S>
__device__ __forceinline__ void apply_ry(float* re, float* im, float c, float s) {
#pragma unroll
  for (int idx = 0; idx < 16; ++idx) {
    if ((idx & S) == 0) {
      const int j = idx | S;
      float r0 = re[idx], i0 = im[idx], r1 = re[j], i1 = im[j];
      re[idx] = c * r0 - s * r1;
      im[idx] = c * i0 - s * i1;
      re[j]   = s * r0 + c * r1;
      im[j]   = s * i0 + c * i1;
    }
  }
}

template<int S>
__device__ __forceinline__ void apply_rz(float* re, float* im, float c, float s) {
#pragma unroll
  for (int idx = 0; idx < 16; ++idx) {
    if ((idx & S) == 0) {
      const int j = idx | S;
      float r0 = re[idx], i0 = im[idx], r1 = re[j], i1 = im[j];
      // |0> amp *= e^{-i t/2};  |1> amp *= e^{+i t/2}
      re[idx] = c * r0 + s * i0;
      im[idx] = c * i0 - s * r0;
      re[j]   = c * r1 - s * i1;
      im[j]   = c * i1 + s * r1;
    }
  }
}

template<int CS, int TS>
__device__ __forceinline__ void apply_cnot(float* re, float* im) {
#pragma unroll
  for (int idx = 0; idx < 16; ++idx) {
    if ((idx & CS) && !(idx & TS)) {
      const int j = idx | TS;
      float tr = re[idx]; re[idx] = re[j]; re[j] = tr;
      float ti = im[idx]; im[idx] = im[j]; im[j] = ti;
    }
  }
}

// ---------------- kernel 1: quanvolution (one thread per patch) ----------------
__global__ void __launch_bounds__(256)
quanv_kernel(const float* __restrict__ x, const float* __restrict__ q,
             float* __restrict__ feats) {
  const int tid = blockIdx.x * blockDim.x + threadIdx.x;
  if (tid >= BATCH * NPATCH) return;
  const int b    = tid / NPATCH;
  const int pidx = tid - b * NPATCH;
  const int i    = pidx / NPS;
  const int j    = pidx - i * NPS;

  const float* img = x + b * (IMG * IMG);
  const float2 top = *(const float2*)&img[(2 * i)     * IMG + 2 * j];
  const float2 bot = *(const float2*)&img[(2 * i + 1) * IMG + 2 * j];

  // RY(p)|0> per qubit: (cos(p/2), sin(p/2))
  float pc[4], ps[4];
  pc[0] = cosf(0.5f * top.x); ps[0] = sinf(0.5f * top.x);
  pc[1] = cosf(0.5f * top.y); ps[1] = sinf(0.5f * top.y);
  pc[2] = cosf(0.5f * bot.x); ps[2] = sinf(0.5f * bot.x);
  pc[3] = cosf(0.5f * bot.y); ps[3] = sinf(0.5f * bot.y);

  float re[16], im[16];
#pragma unroll
  for (int idx = 0; idx < 16; ++idx) {
    re[idx] = ((idx & 8) ? ps[0] : pc[0]) * ((idx & 4) ? ps[1] : pc[1]) *
              ((idx & 2) ? ps[2] : pc[2]) * ((idx & 1) ? ps[3] : pc[3]);
    im[idx] = 0.0f;
  }

  // circuit params (uniform across threads, hits cache)
  float gc[6], gs[6];
#pragma unroll
  for (int k = 0; k < 6; ++k) {
    gc[k] = cosf(0.5f * q[k]);
    gs[k] = sinf(0.5f * q[k]);
  }

  // OPS: rx w0, ry w1, rz w2, cnot(0,1), rx w3, cnot(2,3), ry w0, rz w1
  apply_rx<8>(re, im, gc[0], gs[0]);
  apply_ry<4>(re, im, gc[1], gs[1]);
  apply_rz<2>(re, im, gc[2], gs[2]);
  apply_cnot<8, 4>(re, im);
  apply_rx<1>(re, im, gc[3], gs[3]);
  apply_cnot<2, 1>(re, im);
  apply_ry<8>(re, im, gc[4], gs[4]);
  apply_rz<4>(re, im, gc[5], gs[5]);

  float z0 = 0.f, z1 = 0.f, z2 = 0.f, z3 = 0.f;
#pragma unroll
  for (int idx = 0; idx < 16; ++idx) {
    const float p = re[idx] * re[idx] + im[idx] * im[idx];
    z0 += (idx & 8) ? -p : p;
    z1 += (idx & 4) ? -p : p;
    z2 += (idx & 2) ? -p : p;
    z3 += (idx & 1) ? -p : p;
  }

  float4 o = make_float4(z0, z1, z2, z3);
  *(float4*)&feats[b * FEAT + pidx * 4] = o;
}

// ---------------- kernel 2: logits via V_WMMA_F32_16X16X4_F32 ----------------
// Grid = (4 M-tiles) x (KSPLIT K-chunks); one wave each. Per chunk: 400 K in
// steps of 4 per WMMA, dual accumulators to halve the RAW dependency chain.
// Per-lane float2 loads land exactly in the 16x4 A / 4x16 B VGPR layouts
// (lanes 0-15 hold K+0,K+1; lanes 16-31 hold K+2,K+3). N=2 is padded to 16 by
// masking columns >= 2 with one pk_mul. Partial 16x2 results go to workspace;
// the deterministic 25-term reduction happens in kernel 3.
__global__ void __launch_bounds__(32)
gemm_wmma_kernel(const float* __restrict__ feats, const float* __restrict__ W,
                 float* __restrict__ partials) {
  const int lane  = threadIdx.x;        // 0..31
  const int mtile = blockIdx.x;         // 0..3
  const int kt    = blockIdx.y;         // 0..KSPLIT-1
  const int mn    = lane & 15;          // A: row-in-tile, B: column
  const int koff  = (lane >> 4) << 1;   // lanes 16-31 hold K+2,K+3
  const int kbase = kt * KCHUNK;

  const float* arow = feats + (size_t)(mtile * 16 + mn) * FEAT + kbase + koff;
  const float* brow = W + (size_t)((mn < 2) ? mn : 0) * FEAT + kbase + koff;
  const float  bmask = (mn < 2) ? 1.0f : 0.0f;

  v8f acc0 = {};
  v8f acc1 = {};
  for (int k = 0; k < KCHUNK; k += 8) {
    v2f a0 = *(const v2f*)(arow + k);
    v2f b0 = *(const v2f*)(brow + k);
    b0 *= bmask;
    acc0 = __builtin_amdgcn_wmma_f32_16x16x4_f32(
        false, a0, false, b0, (short)0, acc0, false, false);

    v2f a1 = *(const v2f*)(arow + k + 4);
    v2f b1 = *(const v2f*)(brow + k + 4);
    b1 *= bmask;
    acc1 = __builtin_amdgcn_wmma_f32_16x16x4_f32(
        false, a1, false, b1, (short)0, acc1, false, false);
  }
  v8f acc = acc0 + acc1;

  // C/D layout: VGPR r -> (M = r + 8*(lane>=16), N = lane&15)
  if (mn < 2) {
    const int rowbase = (lane >> 4) << 3;
#pragma unroll
    for (int r = 0; r < 8; ++r)
      partials[(((mtile * KSPLIT + kt) * 16) + rowbase + r) * 2 + mn] = acc[r];
  }
}

// ---------------- kernel 3: K-split reduction + bias + 2-way log_softmax ------
__global__ void __launch_bounds__(64)
reduce_logsoftmax_kernel(const float* __restrict__ partials,
                         const float* __restrict__ bias,
                         float* __restrict__ out) {
  const int t = threadIdx.x;
  if (t < BATCH) {
    const int mtile = t >> 4;
    const int r     = t & 15;
    float l0 = bias[0];
    float l1 = bias[1];
    for (int kt = 0; kt < KSPLIT; ++kt) {   // fixed order -> deterministic
      const float* p = partials + (((mtile * KSPLIT + kt) * 16) + r) * 2;
      l0 += p[0];
      l1 += p[1];
    }
    const float m   = fmaxf(l0, l1);
    const float lse = m + logf(expf(l0 - m) + expf(l1 - m));
    out[t * 2 + 0] = l0 - lse;
    out[t * 2 + 1] = l1 - lse;
  }
}

// ---------------- launcher ----------------
extern "C" void kernel_launch(void* const* d_in, const int* in_sizes, int n_in,
                              void* d_out, int out_size, void* d_ws, size_t ws_size,
                              hipStream_t stream) {
  const float* x = (const float*)d_in[0];       // [64,100,100]
  const float* W = (const float*)d_in[1];       // [2,10000]
  const float* b = (const float*)d_in[2];       // [2]
  const float* q = (const float*)d_in[3];       // [6]

  float* feats    = (float*)d_ws;                  // [64,10000] = 2.56 MB
  float* partials = feats + (size_t)BATCH * FEAT;  // [4*KSPLIT*16*2] = 3200 floats

  const int npatch_total = BATCH * NPATCH;      // 160000
  quanv_kernel<<<(npatch_total + 255) / 256, 256, 0, stream>>>(x, q, feats);
  gemm_wmma_kernel<<<dim3(4, KSPLIT), 32, 0, stream>>>(feats, W, partials);
  reduce_logsoftmax_kernel<<<1, 64, 0, stream>>>(partials, b, (float*)d_out);
}